// QBNNGameAI_59004260712636
// MI455X (gfx1250) — compile-verified
//
#include <hip/hip_runtime.h>
#include <math.h>

typedef __attribute__((ext_vector_type(2))) float v2f;
typedef __attribute__((ext_vector_type(8))) float v8f;

#define LMIN   0.2f
#define LMAX   0.5f
#define DYNF   0.5f
#define LNEPS  1e-5f
#define AS     66          // activation row stride in floats (16 rows per wave tile)
#define WAVES  3           // waves per block (96 threads)
#define MAXBLOCKS 4096     // persistent blocks; amortize LDS weight staging

// Branch-free lane-XOR butterfly add within each 16-lane half (wave32).
#define SWZ_XOR(x, mask)                                                     \
    __int_as_float(__builtin_amdgcn_ds_swizzle(__float_as_int(x),            \
                                               (((mask) << 10) | 0x1f)))

__device__ __forceinline__ float xorsum16(float v) {
    v += SWZ_XOR(v, 1);
    v += SWZ_XOR(v, 2);
    v += SWZ_XOR(v, 4);
    v += SWZ_XOR(v, 8);
    return v;
}

// Branch-free tanh: copysign(1 - 2/(exp2(2*log2e*|x|) + 1), x)
__device__ __forceinline__ float tanh_f(float x) {
    const float e = __builtin_amdgcn_exp2f(fabsf(x) * 2.8853900817779268f);
    const float r = __builtin_amdgcn_rcpf(e + 1.0f);
    return __builtin_copysignf(fmaf(-2.0f, r, 1.0f), x);
}

// Exact-GELU via branch-free rational erf (Abramowitz-Stegun 7.1.26).
__device__ __forceinline__ float gelu_f(float x) {
    const float z = fabsf(x) * 0.70710678118654752f;
    const float t = __builtin_amdgcn_rcpf(fmaf(0.3275911f, z, 1.0f));
    float p = fmaf(1.061405429f, t, -1.453152027f);
    p = fmaf(p, t, 1.421413741f);
    p = fmaf(p, t, -0.284496736f);
    p = fmaf(p, t, 0.254829592f);
    p = p * t;
    const float e = __builtin_amdgcn_exp2f(-z * z * 1.4426950408889634f);
    const float erfa = fmaf(-p, e, 1.0f);
    return 0.5f * x * (1.0f + __builtin_copysignf(erfa, x));
}

__device__ __forceinline__ float sigm(float x) {
    return __builtin_amdgcn_rcpf(
        1.0f + __builtin_amdgcn_exp2f(-x * 1.4426950408889634f));
}

// One QBNN layer on a 16-row tile in this wave's LDS buffer `act`.
// All operand arrays are zero-padded in LDS -> every load is unconditional.
// W: [n][k] stride WS; J: [k][n] stride JS. NORM_N = true feature count.
template <int KSTEPS, int NTILES, int WS, int JS, int NORM_N>
__device__ __forceinline__ void qbnn_layer_tile(
    float* act, int m, int hi,
    const float* __restrict__ W, const float* __restrict__ J,
    const float* __restrict__ b, const float* __restrict__ g,
    const float* __restrict__ be, float lam)
{
    v8f accW[NTILES], accJ[NTILES];
#pragma unroll
    for (int t = 0; t < NTILES; ++t) { accW[t] = {}; accJ[t] = {}; }

#pragma unroll
    for (int ks = 0; ks < KSTEPS; ++ks) {
        const int k0 = ks * 4 + 2 * hi;           // A/B k-slot for this lane half
        const v2f a = *(const v2f*)&act[m * AS + k0];        // ds_load_b64
        v2f aS; aS.x = tanh_f(a.x); aS.y = tanh_f(a.y);      // s_prev frag
#pragma unroll
        for (int t = 0; t < NTILES; ++t) {
            const int n = t * 16 + m;
            const v2f bW = *(const v2f*)&W[n * WS + k0];     // ds_load_b64
            v2f bJ;
            bJ.x = J[(k0)     * JS + n];                     // ds_load_2addr
            bJ.y = J[(k0 + 1) * JS + n];
            accW[t] = __builtin_amdgcn_wmma_f32_16x16x4_f32(
                false, a, false, bW, (short)0, accW[t], false, false);
            accJ[t] = __builtin_amdgcn_wmma_f32_16x16x4_f32(
                false, aS, false, bJ, (short)0, accJ[t], false, false);
        }
    }

    // Fused elementwise in C layout: h_hat = h~ + lam*(sPrev@J)*tanh(h~).
    // Padded cols have W/J/b == 0 -> h_hat == 0 automatically.
#pragma unroll
    for (int t = 0; t < NTILES; ++t) {
        const int col = t * 16 + m;
        const float bc = b[col];
#pragma unroll
        for (int v = 0; v < 8; ++v) {
            const float ht = accW[t][v] + bc;
            const float sr = tanh_f(ht);
            accW[t][v] = ht + lam * (accJ[t][v] * sr);
        }
    }

    // In-register LayerNorm, one-pass stats (E[x^2] - mu^2); padded cols
    // contribute zero to both sums. XOR-butterfly across 16-lane halves.
    float mu[8], rstd[8];
#pragma unroll
    for (int v = 0; v < 8; ++v) {
        float s = 0.0f, q = 0.0f;
#pragma unroll
        for (int t = 0; t < NTILES; ++t) {
            const float h = accW[t][v];
            s += h;
            q = fmaf(h, h, q);
        }
        s = xorsum16(s);
        q = xorsum16(q);
        const float mv = s * (1.0f / (float)NORM_N);
        mu[v] = mv;
        rstd[v] = __builtin_amdgcn_rsqf(
            fmaf(-mv, mv, q * (1.0f / (float)NORM_N)) + LNEPS);
    }

    // Normalize + GELU, write next layer's input tile to LDS.
    // Padded cols: (0-mu)*rstd*0 + 0 = 0, gelu(0) = 0 -> stays zero.
#pragma unroll
    for (int t = 0; t < NTILES; ++t) {
        const int col = t * 16 + m;
        const float gc  = g[col];
        const float bec = be[col];
#pragma unroll
        for (int v = 0; v < 8; ++v) {
            const int row = v + 8 * hi;
            const float y = (accW[t][v] - mu[v]) * rstd[v] * gc + bec;
            act[row * AS + col] = gelu_f(y);
        }
    }
    asm volatile("s_wait_dscnt 0" ::: "memory");
}

__device__ __forceinline__ void zero_lds(float* dst, int n, int tid, int nthr) {
    for (int i = tid; i < n; i += nthr) dst[i] = 0.0f;
}
__device__ __forceinline__ void copy_lds(float* dst, const float* __restrict__ src,
                                         int n, int tid, int nthr) {
    for (int i = tid; i < n; i += nthr) dst[i] = src[i];
}
// Copy src[rows][cs] into zero-padded dst[rows][cd].
__device__ __forceinline__ void copy_pad(float* dst, const float* __restrict__ src,
                                         int rows, int cs, int cd, int tid, int nthr) {
    for (int i = tid; i < rows * cs; i += nthr)
        dst[(i / cs) * cd + (i % cs)] = src[i];
}

__global__ __launch_bounds__(32 * WAVES) void qbnn_fused_kernel(
    const float* __restrict__ x,                                     // [B,9]
    const float* __restrict__ J1, const float* __restrict__ W1,
    const float* __restrict__ b1, const float* __restrict__ be1,
    const float* __restrict__ g1, const float* __restrict__ lb1,
    const float* __restrict__ J2, const float* __restrict__ W2,
    const float* __restrict__ b2, const float* __restrict__ be2,
    const float* __restrict__ g2, const float* __restrict__ lb2,
    const float* __restrict__ J3, const float* __restrict__ W3,
    const float* __restrict__ b3, const float* __restrict__ be3,
    const float* __restrict__ g3, const float* __restrict__ lb3,
    const float* __restrict__ oW, const float* __restrict__ ob,
    float* __restrict__ out, int B, int ntiles)
{
    // Zero-padded LDS weight images (no guards anywhere in the hot loop).
    __shared__ float sW1[64 * 12], sJ1[12 * 64];       // l1: K 9 -> 12
    __shared__ float sW2[64 * 64], sJ2[64 * 64];       // l2: exact
    __shared__ float sW3[16 * 64], sJ3[64 * 16];       // l3: N 9 -> 16
    __shared__ float sOW[16 * 12];                     // head: 9x9 -> 16x12
    __shared__ float sb1[64], sg1[64], sbe1[64];
    __shared__ float sb2[64], sg2[64], sbe2[64];
    __shared__ float sb3[16], sg3[16], sbe3[16], sob[16];
    __shared__ float s_act[WAVES][16 * AS];

    const int tid  = threadIdx.x;
    const int nthr = blockDim.x;

    // Phase 1: zero the padded arrays.
    zero_lds(sW1, 64 * 12, tid, nthr);  zero_lds(sJ1, 12 * 64, tid, nthr);
    zero_lds(sW3, 16 * 64, tid, nthr);  zero_lds(sJ3, 64 * 16, tid, nthr);
    zero_lds(sOW, 16 * 12, tid, nthr);
    zero_lds(sb3, 16, tid, nthr);       zero_lds(sg3, 16, tid, nthr);
    zero_lds(sbe3, 16, tid, nthr);      zero_lds(sob, 16, tid, nthr);
    __syncthreads();

    // Phase 2: fill.
    copy_pad(sW1, W1, 64, 9, 12, tid, nthr);           // [64][9] -> [64][12]
    copy_lds(sJ1, J1, 9 * 64, tid, nthr);              // [9][64] head of [12][64]
    copy_lds(sW2, W2, 64 * 64, tid, nthr);
    copy_lds(sJ2, J2, 64 * 64, tid, nthr);
    copy_lds(sW3, W3, 9 * 64, tid, nthr);              // [9][64] head of [16][64]
    copy_pad(sJ3, J3, 64, 9, 16, tid, nthr);           // [64][9] -> [64][16]
    copy_pad(sOW, oW, 9, 9, 12, tid, nthr);            // [9][9]  -> [16][12]
    copy_lds(sb1, b1, 64, tid, nthr);  copy_lds(sg1, g1, 64, tid, nthr);
    copy_lds(sbe1, be1, 64, tid, nthr);
    copy_lds(sb2, b2, 64, tid, nthr);  copy_lds(sg2, g2, 64, tid, nthr);
    copy_lds(sbe2, be2, 64, tid, nthr);
    copy_lds(sb3, b3, 9, tid, nthr);   copy_lds(sg3, g3, 9, tid, nthr);
    copy_lds(sbe3, be3, 9, tid, nthr); copy_lds(sob, ob, 9, tid, nthr);
    __syncthreads();

    const float lam1 = LMIN + (LMAX - LMIN) * (sigm(lb1[0]) * 0.7f + DYNF * 0.3f);
    const float lam2 = LMIN + (LMAX - LMIN) * (sigm(lb2[0]) * 0.7f + DYNF * 0.3f);
    const float lam3 = LMIN + (LMAX - LMIN) * (sigm(lb3[0]) * 0.7f + DYNF * 0.3f);

    const int lane = tid & 31;
    const int wave = tid >> 5;
    float* act = &s_act[wave][0];
    const int m  = lane & 15;
    const int hi = lane >> 4;
    const long long gN = (long long)B * 9;
    const int tstride = gridDim.x * WAVES;

    for (int tile = blockIdx.x * WAVES + wave; tile < ntiles; tile += tstride) {
        const int row0 = tile * 16;
        const long long base = (long long)row0 * 9;

        // Prefetch next tile's input rows while this tile computes.
        if (tile + tstride < ntiles)
            __builtin_prefetch(&x[(long long)(tile + tstride) * 144 + lane * 4],
                               0, 1);

        // Stage 16x9 input rows: flat coalesced loads, scatter into tile.
#pragma unroll
        for (int it = 0; it < 5; ++it) {
            const int i = it * 32 + lane;
            if (i < 144) {
                const long long gi = base + i;
                const float v = (gi < gN) ? x[gi] : 0.0f;
                act[(i / 9) * AS + (i % 9)] = v;
            }
        }
        // Zero pad cols 9..11 (l1 reads K=12).
#pragma unroll
        for (int it = 0; it < 2; ++it) {
            const int i = it * 32 + lane;
            if (i < 48) act[(i / 3) * AS + 9 + (i % 3)] = 0.0f;
        }
        asm volatile("s_wait_dscnt 0" ::: "memory");

        qbnn_layer_tile<3, 4, 12, 64, 64>(act, m, hi, sW1, sJ1, sb1, sg1, sbe1, lam1);
        qbnn_layer_tile<16, 4, 64, 64, 64>(act, m, hi, sW2, sJ2, sb2, sg2, sbe2, lam2);
        qbnn_layer_tile<16, 1, 64, 16, 9>(act, m, hi, sW3, sJ3, sb3, sg3, sbe3, lam3);

        // Output head: y = h3 @ oW^T + ob (9x9, K padded to 12, N to 16).
        v8f acc = {};
#pragma unroll
        for (int ks = 0; ks < 3; ++ks) {
            const int k0 = ks * 4 + 2 * hi;
            const v2f a  = *(const v2f*)&act[m * AS + k0];
            const v2f bv = *(const v2f*)&sOW[m * 12 + k0];
            acc = __builtin_amdgcn_wmma_f32_16x16x4_f32(
                false, a, false, bv, (short)0, acc, false, false);
        }
        // Write head tile (+bias) back to LDS, then flat coalesced stores.
        {
            const float bc = sob[m];
#pragma unroll
            for (int v = 0; v < 8; ++v)
                act[(v + 8 * hi) * AS + m] = acc[v] + bc;
        }
        asm volatile("s_wait_dscnt 0" ::: "memory");
#pragma unroll
        for (int it = 0; it < 5; ++it) {
            const int i = it * 32 + lane;
            if (i < 144) {
                const long long gi = base + i;
                if (gi < gN) out[gi] = act[(i / 9) * AS + (i % 9)];
            }
        }
    }
}

extern "C" void kernel_launch(void* const* d_in, const int* in_sizes, int n_in,
                              void* d_out, int out_size, void* d_ws, size_t ws_size,
                              hipStream_t stream) {
    // Input order: JAX pytree flatten (dict keys sorted alphabetically):
    // 0: board_state; per layer (l1,l2,l3): J, W, b, beta, gamma, lambda_base;
    // 19: out_W; 20: out_b
    const float* x   = (const float*)d_in[0];
    const float* J1  = (const float*)d_in[1];
    const float* W1  = (const float*)d_in[2];
    const float* b1  = (const float*)d_in[3];
    const float* be1 = (const float*)d_in[4];
    const float* g1  = (const float*)d_in[5];
    const float* lb1 = (const float*)d_in[6];
    const float* J2  = (const float*)d_in[7];
    const float* W2  = (const float*)d_in[8];
    const float* b2  = (const float*)d_in[9];
    const float* be2 = (const float*)d_in[10];
    const float* g2  = (const float*)d_in[11];
    const float* lb2 = (const float*)d_in[12];
    const float* J3  = (const float*)d_in[13];
    const float* W3  = (const float*)d_in[14];
    const float* b3  = (const float*)d_in[15];
    const float* be3 = (const float*)d_in[16];
    const float* g3  = (const float*)d_in[17];
    const float* lb3 = (const float*)d_in[18];
    const float* oW  = (const float*)d_in[19];
    const float* ob  = (const float*)d_in[20];
    float* out = (float*)d_out;

    const int B      = in_sizes[0] / 9;
    const int ntiles = (B + 15) / 16;
    int blocks = (ntiles + WAVES - 1) / WAVES;
    if (blocks > MAXBLOCKS) blocks = MAXBLOCKS;   // persistent blocks

    qbnn_fused_kernel<<<blocks, 32 * WAVES, 0, stream>>>(
        x, J1, W1, b1, be1, g1, lb1,
        J2, W2, b2, be2, g2, lb2,
        J3, W3, b3, be3, g3, lb3,
        oW, ob, out, B, ntiles);
}